// GaussianRender_53231824666914
// MI455X (gfx1250) — compile-verified
//
#include <hip/hip_runtime.h>

typedef __attribute__((ext_vector_type(2))) float v2f;
typedef __attribute__((ext_vector_type(8))) float v8f;

#define TSZ   64
#define KG    256
#define WPIX  512

// lane <-> lane^16 exchange within wave32 (SWAPX16): xor=0x10, or=0, and=0x1f
__device__ __forceinline__ float swapx16(float x) {
  return __int_as_float(__builtin_amdgcn_ds_swizzle(__float_as_int(x), 0x401F));
}

__global__ __launch_bounds__(256) void GaussianRender_kernel(
    const float* __restrict__ mu,    // [N,2]
    const float* __restrict__ cov,   // [N,2,2]
    const float* __restrict__ opac,  // [N]
    const float* __restrict__ color, // [N,3]
    const int*   __restrict__ tidx,  // [T,K]
    float*       __restrict__ out)   // [H,W,3]
{
  __shared__ __align__(16) float  feat[KG * 8];  // 8KB: scaled rank-6 features + 2 zero pads
  __shared__ __align__(16) float4 opcol[KG];     // 4KB: {opacity, r, g, b}

  const int blk = blockIdx.x;
  const int t   = blk >> 5;        // tile 0..63
  const int sub = blk & 31;        // 128-pixel slab within tile
  const int ty  = t >> 3, tx = t & 7;

  // ---- preprocess this tile's 256 gaussians into LDS (1 per thread) ----
  {
    const int k = threadIdx.x;
    const int n = tidx[t * KG + k];
    const float a = cov[n*4+0], b = cov[n*4+1], c = cov[n*4+2], d = cov[n*4+3];
    const float det = fmaxf(a*d - b*c, 1e-6f);
    const float ia  =  d / det;
    const float ibc = (-b / det) + (-c / det);
    const float id  =  a / det;
    const float mx = mu[n*2+0] - (float)(tx * TSZ);   // tile-local center
    const float my = mu[n*2+1] - (float)(ty * TSZ);
    const float s  = -0.72134752044448170f;           // -0.5 * log2(e)
    // maha expanded in monomials [x^2, xy, y^2, x, y, 1], pre-scaled so exp2(D)=exp(-maha/2)
    feat[k*8+0] = s * ia;
    feat[k*8+1] = s * ibc;
    feat[k*8+2] = s * id;
    feat[k*8+3] = -s * (2.0f*ia*mx + ibc*my);
    feat[k*8+4] = -s * (2.0f*id*my + ibc*mx);
    feat[k*8+5] = s * (ia*mx*mx + ibc*mx*my + id*my*my);
    feat[k*8+6] = 0.0f;                                // K=6,7 pad (read by hi lanes of 2nd WMMA)
    feat[k*8+7] = 0.0f;
    opcol[k] = make_float4(opac[n], color[n*3+0], color[n*3+1], color[n*3+2]);
  }
  __syncthreads();

  const int  lane = threadIdx.x & 31;
  const int  wav  = threadIdx.x >> 5;
  const bool hi   = lane >= 16;
  const int  p    = lane & 15;                 // pixel column this lane owns

  const int   pix = sub * 128 + wav * 16 + p;  // 0..4095 within tile
  const float px  = (float)(pix & 63) + 0.5f;
  const float py  = (float)(pix >> 6) + 0.5f;

  // B operand (4xK x 16 pixels): vgpr0 = K0(lo)/K2(hi), vgpr1 = K1(lo)/K3(hi)
  v2f B0, B1;
  B0.x = hi ? py*py : px*px;   // K2 : K0
  B0.y = hi ? px    : px*py;   // K3 : K1
  B1.x = hi ? 0.0f  : py;      // K6 : K4
  B1.y = hi ? 0.0f  : 1.0f;    // K7 : K5

  const float* fbase = feat + (hi ? 2 : 0);    // A: lo lanes feed K0,1 / hi lanes K2,3
  float S = 0.0f, accR = 0.0f, accG = 0.0f, accB = 0.0f;

  for (int c = 0; c < 16; ++c) {
    const int base = c * 16;
    const int ga   = base + p;                 // A-matrix row = gaussian index
    const v2f a0 = *(const v2f*)(fbase + ga*8);       // feats {0,1} / {2,3}
    const v2f a1 = *(const v2f*)(fbase + ga*8 + 4);   // feats {4,5} / {6,7}=0

    // D[v] = -0.5*log2e * maha for gaussian (base + v + 8*hi) at pixel p
    v8f D = {};
    D = __builtin_amdgcn_wmma_f32_16x16x4_f32(false, a0, false, B0, (short)0, D, false, false);
    D = __builtin_amdgcn_wmma_f32_16x16x4_f32(false, a1, false, B1, (short)0, D, false, false);

    const int gofs = base + (hi ? 8 : 0);
    float4 oc[8];
    float  alpha[8];
#pragma unroll
    for (int v = 0; v < 8; ++v) oc[v] = opcol[gofs + v];
#pragma unroll
    for (int v = 0; v < 8; ++v) {
      const float prob = __builtin_amdgcn_exp2f(D[v]);
      alpha[v] = fminf(fmaxf(oc[v].x * prob, 0.01f), 0.99f);
    }
    float ssum = 0.0f;
#pragma unroll
    for (int v = 0; v < 8; ++v) ssum += alpha[v];

    const float other = swapx16(ssum);         // other half's alpha sum
    float pre = S + (hi ? other : 0.0f);       // hi lanes start after gaussians 0-7
#pragma unroll
    for (int v = 0; v < 8; ++v) {
      const float w = alpha[v] * (1.0f - pre); // trans_k = 1 - sum_{j<k} alpha_j
      pre += alpha[v];
      accR = fmaf(w, oc[v].y, accR);
      accG = fmaf(w, oc[v].z, accG);
      accB = fmaf(w, oc[v].w, accB);
    }
    S += ssum + other;                         // identical in both halves
  }

  // merge the two gaussian-halves of each pixel
  accR += swapx16(accR);
  accG += swapx16(accG);
  accB += swapx16(accB);

  if (!hi) {
    const int lx = pix & 63, ly = pix >> 6;
    const int gx = tx * TSZ + lx, gy = ty * TSZ + ly;
    float* o = out + ((size_t)gy * WPIX + gx) * 3;
    o[0] = accR; o[1] = accG; o[2] = accB;
  }
}

extern "C" void kernel_launch(void* const* d_in, const int* in_sizes, int n_in,
                              void* d_out, int out_size, void* d_ws, size_t ws_size,
                              hipStream_t stream) {
  (void)in_sizes; (void)n_in; (void)d_ws; (void)ws_size; (void)out_size;
  const float* mu    = (const float*)d_in[0];
  const float* cov   = (const float*)d_in[1];
  const float* opac  = (const float*)d_in[2];
  const float* color = (const float*)d_in[3];
  const int*   tidx  = (const int*)d_in[4];
  float* out = (float*)d_out;
  // 64 tiles * 32 slabs of 128 pixels; 256 threads = 8 wave32 per block
  GaussianRender_kernel<<<dim3(64 * 32), dim3(256), 0, stream>>>(
      mu, cov, opac, color, tidx, out);
}